// SDFMarcher_46145128628269
// MI455X (gfx1250) — compile-verified
//
#include <hip/hip_runtime.h>

// CDNA5 (gfx1250) SDF sphere-marcher: fp32 WMMA (v_wmma_f32_16x16x4_f32) for
// the 128x128 hidden layer. Accumulator-resident / operand-streaming form:
// 8 C/D tiles live in VGPRs, A (layer-1 activations) and B (W2 from LDS)
// streamed per K-chunk. waves_per_eu>=2 keeps allocation under 256 VGPRs so
// the 8-wave workgroup co-schedules 2 waves/SIMD and avoids s_set_vgpr_msb.

typedef float v2f __attribute__((ext_vector_type(2)));
typedef float v8f __attribute__((ext_vector_type(8)));

#define NRAYS 65536
#define HID   128
#define WAVES_PER_BLOCK 8
#define THREADS (WAVES_PER_BLOCK * 32)

__global__ __launch_bounds__(THREADS, 2)
void sdf_march_wmma(const float* __restrict__ P,   // [N,3] positions
                    const float* __restrict__ Dir, // [N,3] directions
                    const float* __restrict__ W1,  // [3,128]
                    const float* __restrict__ B1,  // [128]
                    const float* __restrict__ W2,  // [128,128]
                    const float* __restrict__ B2,  // [128]
                    const float* __restrict__ W3,  // [128,1]
                    const float* __restrict__ B3,  // [1]
                    const int*   __restrict__ STEPS,
                    float* __restrict__ OUT)       // [N,3]
{
    // W2 staged in WMMA fp32 B-matrix layout:
    //   entry e = (n*32 + kk)*32 + lane, lane l: {W2[4kk+2*(l>>4)][16n+(l&15)],
    //                                             W2[4kk+2*(l>>4)+1][16n+(l&15)]}
    __shared__ float2 w2s[8 * 32 * 32];   // 64 KB
    __shared__ float2 w1v[3 * 64];        // W1 as float2 pairs along j
    __shared__ float2 b1v[64];

    const int t = threadIdx.x;

    // ---- stage weights into LDS (once per block; W2 is L2-resident) ----
    for (int e = t; e < 8 * 32 * 32; e += THREADS) {
        const int l  = e & 31;
        const int kk = (e >> 5) & 31;
        const int n  = e >> 10;
        const int col = 16 * n + (l & 15);
        const int k0  = 4 * kk + 2 * (l >> 4);
        float2 w;
        w.x = W2[k0 * HID + col];
        w.y = W2[(k0 + 1) * HID + col];
        w2s[e] = w;
    }
    for (int e = t; e < 3 * 64; e += THREADS) w1v[e] = ((const float2*)W1)[e];
    for (int e = t; e < 64; e += THREADS)     b1v[e] = ((const float2*)B1)[e];
    __syncthreads();

    const int lane = t & 31;
    const int hi   = lane >> 4;   // half-wave select (K-offset / row-offset)
    const int lo   = lane & 15;   // ray-in-tile == column-in-tile
    const int wave = t >> 5;
    const int ray  = (blockIdx.x * WAVES_PER_BLOCK + wave) * 16 + lo;

    // Lane l holds ray (l & 15)'s state, replicated across half-waves.
    float px = P[3 * ray + 0], py = P[3 * ray + 1], pz = P[3 * ray + 2];
    const float dx = Dir[3 * ray + 0], dy = Dir[3 * ray + 1], dz = Dir[3 * ray + 2];
    const float b3 = B3[0];
    const int steps = STEPS[0];

    // Per-lane constants for the C-bias seed and the layer-3 fold.
    // C layout: column = lane%16, identical for all 8 row-VGPRs -> splat.
    float biasv[8], w3cv[8];
#pragma unroll
    for (int n = 0; n < 8; ++n) {
        biasv[n] = B2[16 * n + lo];
        w3cv[n]  = W3[16 * n + lo];
    }

#pragma unroll 1
    for (int s = 0; s < steps; ++s) {
        // Opaque zero: keeps the per-step LDS streams inside the loop
        // (otherwise LICM hoists 500+ VGPRs of loop-invariant loads -> spills).
        int off = 0;
        asm volatile("" : "+v"(off));

        // ---- seed the 8 accumulator tiles with b2 ----
        v8f acc[8];
#pragma unroll
        for (int n = 0; n < 8; ++n) {
            const float b = biasv[n];
            acc[n] = (v8f){b, b, b, b, b, b, b, b};
        }

        // ---- fused layer1 + layer2: stream A (VALU) and B (LDS) per K-chunk ----
        // A layout (16x4 f32): lane l<16 rows M=l K%4 in {0,1}; lanes 16..31 K%4 in {2,3}
#pragma unroll
        for (int kk = 0; kk < 32; ++kk) {
            const int idx = 2 * kk + hi + off;        // j0 = 4kk + 2hi, j1 = j0+1
            const float2 r0 = w1v[idx];
            const float2 r1 = w1v[64 + idx];
            const float2 r2 = w1v[128 + idx];
            const float2 bb = b1v[idx];
            const float h0 = fmaf(pz, r2.x, fmaf(py, r1.x, fmaf(px, r0.x, bb.x)));
            const float h1 = fmaf(pz, r2.y, fmaf(py, r1.y, fmaf(px, r0.y, bb.y)));
            v2f av; av.x = fmaxf(h0, 0.0f); av.y = fmaxf(h1, 0.0f);

#pragma unroll
            for (int n = 0; n < 8; ++n) {
                const float2 bw = w2s[(n * 32 + kk) * 32 + lane + off];
                v2f bv; bv.x = bw.x; bv.y = bw.y;
                // D = A(16x4) * B(4x16) + C   (v_wmma_f32_16x16x4_f32)
                acc[n] = __builtin_amdgcn_wmma_f32_16x16x4_f32(
                    /*neg_a=*/false, av, /*neg_b=*/false, bv,
                    /*c_mod=*/(short)0, acc[n], /*reuse_a=*/false, /*reuse_b=*/false);
            }
        }

        // ---- layer 3 folded into the accumulator drain ----
        float part[8];
#pragma unroll
        for (int v = 0; v < 8; ++v) part[v] = 0.0f;
#pragma unroll
        for (int n = 0; n < 8; ++n) {
            const float w3c = w3cv[n];
#pragma unroll
            for (int v = 0; v < 8; ++v)
                part[v] = fmaf(fmaxf(acc[n][v], 0.0f), w3c, part[v]);
        }

        // ---- reduce over the 16 columns within each half-wave ----
#pragma unroll
        for (int m = 1; m <= 8; m <<= 1) {
#pragma unroll
            for (int v = 0; v < 8; ++v)
                part[v] += __shfl_xor(part[v], m, 32);
        }
        // lane picks s for ray (8*hi + lane&7), then cross-half route so each
        // lane gets s[ray = lane&15]
        float sv = part[0];
#pragma unroll
        for (int v = 1; v < 8; ++v)
            sv = ((lane & 7) == v) ? part[v] : sv;
        sv += b3;
        sv = __shfl(sv, 16 * ((lane >> 3) & 1) + (lane & 7), 32);

        // ---- march: p += sdf(p) * dir ----
        px = fmaf(sv, dx, px);
        py = fmaf(sv, dy, py);
        pz = fmaf(sv, dz, pz);
    }

    if (hi == 0) {
        OUT[3 * ray + 0] = px;
        OUT[3 * ray + 1] = py;
        OUT[3 * ray + 2] = pz;
    }
}

extern "C" void kernel_launch(void* const* d_in, const int* in_sizes, int n_in,
                              void* d_out, int out_size, void* d_ws, size_t ws_size,
                              hipStream_t stream) {
    const float* P   = (const float*)d_in[0];
    const float* Dir = (const float*)d_in[1];
    const float* W1  = (const float*)d_in[2];
    const float* B1  = (const float*)d_in[3];
    const float* W2  = (const float*)d_in[4];
    const float* B2  = (const float*)d_in[5];
    const float* W3  = (const float*)d_in[6];
    const float* B3  = (const float*)d_in[7];
    const int*   ST  = (const int*)d_in[8];
    float* OUT = (float*)d_out;

    dim3 grid(NRAYS / (WAVES_PER_BLOCK * 16));   // 512 blocks x 256 threads
    sdf_march_wmma<<<grid, THREADS, 0, stream>>>(P, Dir, W1, B1, W2, B2, W3, B3, ST, OUT);
}